// GraphNeuralNetwork_29918742184185
// MI455X (gfx1250) — compile-verified
//
#include <hip/hip_runtime.h>
#include <hip/hip_bf16.h>

// ---------------- problem constants (match reference) ----------------
#define NN      50000
#define EE      1600000
#define DIN     128
#define EMBD    32
#define KIN     160          // DIN + EMBD
#define HDIM    256
#define ODIM    128
#define NLAYERS 3
#define NB      8
#define LN_EPS  1e-5f

typedef float v2f __attribute__((ext_vector_type(2)));
typedef float v8f __attribute__((ext_vector_type(8)));

// ---------------- fp32 WMMA primitive ----------------
__device__ __forceinline__ v8f wmma4(v2f a, v2f b, v8f c) {
  // D = A(16x4) * B(4x16) + C, full fp32
  return __builtin_amdgcn_wmma_f32_16x16x4_f32(
      /*neg_a=*/false, a, /*neg_b=*/false, b,
      /*c_mod=*/(short)0, c, /*reuse_a=*/false, /*reuse_b=*/false);
}

// One wave computes a 16x16 fp32 tile: C = A[16xK] * B[KxN-tile]
// A is in LDS row-major with row stride lda; B is global row-major ldb.
__device__ __forceinline__ v8f gemm_tile(const float* Alds, int lda,
                                         const float* __restrict__ B, int ldb,
                                         int n0, int K) {
  const int lane = threadIdx.x & 31;
  const int hg   = lane >> 4;   // half-group: 0 -> K {0,1}, 1 -> K {2,3}
  const int l    = lane & 15;
  v8f c = {};
#pragma unroll 8
  for (int k = 0; k < K; k += 4) {
    v2f a, b;
    const float* ap = Alds + l * lda + k + 2 * hg;
    a.x = ap[0];
    a.y = ap[1];
    const float* bp = B + (size_t)(k + 2 * hg) * ldb + n0 + l;
    b.x = bp[0];
    b.y = bp[ldb];
    c = wmma4(a, b, c);
  }
  return c;
}

__device__ __forceinline__ void atomAddF(float* p, float v) {
  __hip_atomic_fetch_add(p, v, __ATOMIC_RELAXED, __HIP_MEMORY_SCOPE_AGENT);
}

// ---------------- kernel 0: concat(x, emb[type]) @ W_in + b_in, relu ----------------
__global__ __launch_bounds__(256) void k_input_gemm(
    const float* __restrict__ x, const int* __restrict__ ntype,
    const float* __restrict__ emb, const float* __restrict__ Win,
    const float* __restrict__ bin, float* __restrict__ h) {
  __shared__ float Atile[16 * (KIN + 4)];
  const int lda  = KIN + 4;
  const int row0 = blockIdx.x * 16;
  const int tid  = threadIdx.x;
  // stage x part: 16 rows x 128 cols
  for (int i = tid; i < 16 * 32; i += 256) {
    int r = i >> 5, cq = i & 31;
    float4 v = ((const float4*)(x + (size_t)(row0 + r) * DIN))[cq];
    float* p = Atile + r * lda + cq * 4;
    p[0] = v.x; p[1] = v.y; p[2] = v.z; p[3] = v.w;
  }
  // stage embedding part: 16 rows x 32 cols
  for (int i = tid; i < 16 * 8; i += 256) {
    int r = i >> 3, cq = i & 7;
    int t = ntype[row0 + r];
    float4 v = ((const float4*)(emb + (size_t)t * EMBD))[cq];
    float* p = Atile + r * lda + DIN + cq * 4;
    p[0] = v.x; p[1] = v.y; p[2] = v.z; p[3] = v.w;
  }
  __syncthreads();
  const int wave = tid >> 5;
  const int lane = tid & 31, hg = lane >> 4, l = lane & 15;
  for (int t = wave; t < HDIM / 16; t += 8) {
    const int n0 = t * 16;
    v8f c = gemm_tile(Atile, lda, Win, HDIM, n0, KIN);
    const float bv = bin[n0 + l];
#pragma unroll
    for (int r = 0; r < 8; ++r) {
      float v = c[r] + bv;
      h[(size_t)(row0 + r + 8 * hg) * HDIM + n0 + l] = v > 0.f ? v : 0.f;
    }
  }
}

// ---------------- kernel 1: m = h @ W  (256x256) ----------------
__global__ __launch_bounds__(256) void k_hidden_gemm(
    const float* __restrict__ h, const float* __restrict__ W,
    float* __restrict__ m) {
  __shared__ float Atile[16 * (HDIM + 4)];
  const int lda  = HDIM + 4;
  const int row0 = blockIdx.x * 16;
  const int tid  = threadIdx.x;
  for (int i = tid; i < 16 * 64; i += 256) {
    int r = i >> 6, cq = i & 63;
    float4 v = ((const float4*)(h + (size_t)(row0 + r) * HDIM))[cq];
    float* p = Atile + r * lda + cq * 4;
    p[0] = v.x; p[1] = v.y; p[2] = v.z; p[3] = v.w;
  }
  __syncthreads();
  const int wave = tid >> 5;
  const int lane = tid & 31, hg = lane >> 4, l = lane & 15;
  for (int t = wave; t < HDIM / 16; t += 8) {
    const int n0 = t * 16;
    v8f c = gemm_tile(Atile, lda, W, HDIM, n0, HDIM);
#pragma unroll
    for (int r = 0; r < 8; ++r)
      m[(size_t)(row0 + r + 8 * hg) * HDIM + n0 + l] = c[r];
  }
}

// ---------------- graph structure: degrees, scan, CSR fill ----------------
__global__ void k_zero_f(float* __restrict__ p, int n) {
  int i = blockIdx.x * blockDim.x + threadIdx.x;
  if (i < n) p[i] = 0.0f;
}
__global__ void k_zero_i(int* __restrict__ p, int n) {
  int i = blockIdx.x * blockDim.x + threadIdx.x;
  if (i < n) p[i] = 0;
}
__global__ void k_indeg(const int* __restrict__ dst, int* __restrict__ indeg) {
  int e = blockIdx.x * blockDim.x + threadIdx.x;
  if (e < EE) atomicAdd(&indeg[dst[e]], 1);
}
__global__ void k_dinv(const int* __restrict__ indeg, float* __restrict__ dinv) {
  int i = blockIdx.x * blockDim.x + threadIdx.x;
  if (i < NN) dinv[i] = rsqrtf((float)indeg[i] + 1.0f);  // +1 self-loop
}

// single-workgroup exclusive scan of in-degrees -> CSR offsets
__global__ __launch_bounds__(1024) void k_scan(const int* __restrict__ indeg,
                                               int* __restrict__ off) {
  __shared__ int buf[1024];
  __shared__ int carry_s;
  if (threadIdx.x == 0) carry_s = 0;
  __syncthreads();
  for (int base = 0; base < NN; base += 1024) {
    int i = base + (int)threadIdx.x;
    int v = (i < NN) ? indeg[i] : 0;
    buf[threadIdx.x] = v;
    __syncthreads();
#pragma unroll
    for (int o = 1; o < 1024; o <<= 1) {
      int t = ((int)threadIdx.x >= o) ? buf[threadIdx.x - o] : 0;
      __syncthreads();
      buf[threadIdx.x] += t;
      __syncthreads();
    }
    int incl  = buf[threadIdx.x];
    int carry = carry_s;
    if (i < NN) off[i] = carry + incl - v;  // exclusive
    __syncthreads();
    if (threadIdx.x == 1023) carry_s = carry + buf[1023];
    __syncthreads();
  }
  if (threadIdx.x == 0) off[NN] = carry_s;  // == EE
}

__global__ void k_fill(const int* __restrict__ src, const int* __restrict__ dst,
                       const int* __restrict__ off, int* __restrict__ cursor,
                       int* __restrict__ csr_src) {
  int e = blockIdx.x * blockDim.x + threadIdx.x;
  if (e >= EE) return;
  int d = dst[e];
  int p = atomicAdd(&cursor[d], 1);
  csr_src[off[d] + p] = src[e];
}

// ---------------- fused gather + bias + layernorm + relu + residual ----------------
// one wave per node; full 256-wide row held in 8 regs/lane
__global__ __launch_bounds__(256) void k_gather_ln(
    const float* __restrict__ m, const float* __restrict__ dinv,
    const int* __restrict__ off, const int* __restrict__ csr_src,
    const int* __restrict__ indeg,
    const float* __restrict__ bias, const float* __restrict__ gamma,
    const float* __restrict__ beta, const float* __restrict__ hold,
    float* __restrict__ hnew) {
  const int wave = threadIdx.x >> 5, lane = threadIdx.x & 31;
  const int node = blockIdx.x * 8 + wave;
  const float di = dinv[node];

  // self-loop contribution: m[node] * dinv^2
  const float4* mrow = (const float4*)(m + (size_t)node * HDIM);
  float4 a0 = mrow[lane];
  float4 a1 = mrow[lane + 32];
  const float w = di * di;
  a0.x *= w; a0.y *= w; a0.z *= w; a0.w *= w;
  a1.x *= w; a1.y *= w; a1.z *= w; a1.w *= w;

  const int base = off[node];
  const int cnt  = indeg[node];
  for (int j = 0; j < cnt; ++j) {
    const int s = csr_src[base + j];            // wave-uniform
    if (j + 1 < cnt) {
      const int sn = csr_src[base + j + 1];
      __builtin_prefetch(m + (size_t)sn * HDIM + lane * 4, 0, 3);
    }
    const float nrm = dinv[s] * di;
    const float4* sr = (const float4*)(m + (size_t)s * HDIM);
    float4 v0 = sr[lane];
    float4 v1 = sr[lane + 32];
    a0.x += v0.x * nrm; a0.y += v0.y * nrm; a0.z += v0.z * nrm; a0.w += v0.w * nrm;
    a1.x += v1.x * nrm; a1.y += v1.y * nrm; a1.z += v1.z * nrm; a1.w += v1.w * nrm;
  }

  // + bias
  const float4* b4 = (const float4*)bias;
  {
    float4 bb = b4[lane];
    a0.x += bb.x; a0.y += bb.y; a0.z += bb.z; a0.w += bb.w;
  }
  {
    float4 bb = b4[lane + 32];
    a1.x += bb.x; a1.y += bb.y; a1.z += bb.z; a1.w += bb.w;
  }

  // layernorm over 256 channels (wave32 reduction)
  float s = a0.x + a0.y + a0.z + a0.w + a1.x + a1.y + a1.z + a1.w;
#pragma unroll
  for (int o = 16; o > 0; o >>= 1) s += __shfl_xor(s, o, 32);
  const float mu = s * (1.0f / HDIM);
  float q = (a0.x - mu) * (a0.x - mu) + (a0.y - mu) * (a0.y - mu) +
            (a0.z - mu) * (a0.z - mu) + (a0.w - mu) * (a0.w - mu) +
            (a1.x - mu) * (a1.x - mu) + (a1.y - mu) * (a1.y - mu) +
            (a1.z - mu) * (a1.z - mu) + (a1.w - mu) * (a1.w - mu);
#pragma unroll
  for (int o = 16; o > 0; o >>= 1) q += __shfl_xor(q, o, 32);
  const float rstd = rsqrtf(q * (1.0f / HDIM) + LN_EPS);

  const float4* g4 = (const float4*)gamma;
  const float4* be4 = (const float4*)beta;
  const float4* h4 = (const float4*)(hold + (size_t)node * HDIM);
  float4* o4 = (float4*)(hnew + (size_t)node * HDIM);
  {
    float4 g = g4[lane], b = be4[lane], h0 = h4[lane], o;
    o.x = fmaxf((a0.x - mu) * rstd * g.x + b.x, 0.f) + h0.x;
    o.y = fmaxf((a0.y - mu) * rstd * g.y + b.y, 0.f) + h0.y;
    o.z = fmaxf((a0.z - mu) * rstd * g.z + b.z, 0.f) + h0.z;
    o.w = fmaxf((a0.w - mu) * rstd * g.w + b.w, 0.f) + h0.w;
    o4[lane] = o;
  }
  {
    float4 g = g4[lane + 32], b = be4[lane + 32], h0 = h4[lane + 32], o;
    o.x = fmaxf((a1.x - mu) * rstd * g.x + b.x, 0.f) + h0.x;
    o.y = fmaxf((a1.y - mu) * rstd * g.y + b.y, 0.f) + h0.y;
    o.z = fmaxf((a1.z - mu) * rstd * g.z + b.z, 0.f) + h0.z;
    o.w = fmaxf((a1.w - mu) * rstd * g.w + b.w, 0.f) + h0.w;
    o4[lane + 32] = o;
  }
}

// ---------------- output GEMM fused with pooling sums ----------------
__global__ __launch_bounds__(256) void k_out_gemm(
    const float* __restrict__ h, const float* __restrict__ Wout,
    const float* __restrict__ bout, const int* __restrict__ batch,
    float* __restrict__ sums) {
  __shared__ float Atile[16 * (HDIM + 4)];
  const int lda  = HDIM + 4;
  const int row0 = blockIdx.x * 16;
  const int tid  = threadIdx.x;
  for (int i = tid; i < 16 * 64; i += 256) {
    int r = i >> 6, cq = i & 63;
    float4 v = ((const float4*)(h + (size_t)(row0 + r) * HDIM))[cq];
    float* p = Atile + r * lda + cq * 4;
    p[0] = v.x; p[1] = v.y; p[2] = v.z; p[3] = v.w;
  }
  __syncthreads();
  const int wave = tid >> 5;                 // 8 waves -> 8 col tiles of 16
  const int n0   = wave * 16;
  v8f c = gemm_tile(Atile, lda, Wout, ODIM, n0, HDIM);
  const int lane = tid & 31, hg = lane >> 4, l = lane & 15;
  const float bv = bout[n0 + l];
#pragma unroll
  for (int r = 0; r < 8; ++r) {
    int row = row0 + r + 8 * hg;
    atomAddF(&sums[(size_t)batch[row] * ODIM + n0 + l], c[r] + bv);
  }
}

// ---------------- pooling helpers ----------------
__global__ void k_cnt(const int* __restrict__ batch, float* __restrict__ cnt) {
  int i = blockIdx.x * blockDim.x + threadIdx.x;
  if (i < NN) atomAddF(&cnt[batch[i]], 1.0f);
}
__global__ void k_finalize(const float* __restrict__ sums, const float* __restrict__ cnt,
                           float* __restrict__ out) {
  int i = blockIdx.x * blockDim.x + threadIdx.x;
  if (i < NB * ODIM) out[i] = sums[i] / fmaxf(cnt[i >> 7], 1.0f);
}

// ---------------- host driver ----------------
extern "C" void kernel_launch(void* const* d_in, const int* in_sizes, int n_in,
                              void* d_out, int out_size, void* d_ws, size_t ws_size,
                              hipStream_t stream) {
  const float* x      = (const float*)d_in[0];
  const int*   ei     = (const int*)  d_in[1];
  const int*   ntype  = (const int*)  d_in[2];
  const int*   batch  = (const int*)  d_in[3];
  const float* emb    = (const float*)d_in[4];
  const float* Win    = (const float*)d_in[5];
  const float* bin    = (const float*)d_in[6];
  const float* Ws     = (const float*)d_in[7];
  const float* bs     = (const float*)d_in[8];
  const float* gammas = (const float*)d_in[9];
  const float* betas  = (const float*)d_in[10];
  const float* Wout   = (const float*)d_in[11];
  const float* bout   = (const float*)d_in[12];
  float* out = (float*)d_out;
  (void)in_sizes; (void)n_in; (void)out_size; (void)ws_size;

  const int* src = ei;
  const int* dst = ei + EE;

  const size_t S = (size_t)NN * HDIM;
  float* ws   = (float*)d_ws;
  float* bufA = ws;            // h (current)
  float* bufB = ws + S;        // m
  float* bufC = ws + 2 * S;    // h (next)
  float* dinv = ws + 3 * S;    // NN
  float* sums = dinv + NN;     // NB*ODIM
  float* cnt  = sums + (size_t)NB * ODIM;   // NB
  int*   indeg  = (int*)(cnt + NB);         // NN
  int*   off    = indeg + NN;               // NN+1
  int*   cursor = off + NN + 1;             // NN
  int*   csr    = cursor + NN;              // EE

  const int T = 256;
  // graph-structure precompute (same work every call)
  k_zero_f<<<dim3((NB * ODIM + NB + T - 1) / T), T, 0, stream>>>(sums, NB * ODIM + NB);
  k_zero_i<<<dim3((NN + T - 1) / T), T, 0, stream>>>(indeg, NN);
  k_zero_i<<<dim3((NN + T - 1) / T), T, 0, stream>>>(cursor, NN);
  k_indeg <<<dim3((EE + T - 1) / T), T, 0, stream>>>(dst, indeg);
  k_dinv  <<<dim3((NN + T - 1) / T), T, 0, stream>>>(indeg, dinv);
  k_scan  <<<dim3(1), 1024, 0, stream>>>(indeg, off);
  k_fill  <<<dim3((EE + T - 1) / T), T, 0, stream>>>(src, dst, off, cursor, csr);
  k_cnt   <<<dim3((NN + T - 1) / T), T, 0, stream>>>(batch, cnt);

  // input projection
  k_input_gemm<<<dim3(NN / 16), T, 0, stream>>>(x, ntype, emb, Win, bin, bufA);

  float* hcur = bufA;
  float* hnxt = bufC;
  for (int l = 0; l < NLAYERS; ++l) {
    k_hidden_gemm<<<dim3(NN / 16), T, 0, stream>>>(hcur, Ws + (size_t)l * HDIM * HDIM, bufB);
    k_gather_ln  <<<dim3(NN / 8), T, 0, stream>>>(bufB, dinv, off, csr, indeg,
                                                  bs + (size_t)l * HDIM,
                                                  gammas + (size_t)l * HDIM,
                                                  betas + (size_t)l * HDIM,
                                                  hcur, hnxt);
    float* tmp = hcur; hcur = hnxt; hnxt = tmp;
  }

  // output projection fused with pooled sums; then mean
  k_out_gemm<<<dim3(NN / 16), T, 0, stream>>>(hcur, Wout, bout, batch, sums);
  k_finalize<<<dim3((NB * ODIM + T - 1) / T), T, 0, stream>>>(sums, cnt, out);
}